// BahdanauAttention_72103910965793
// MI455X (gfx1250) — compile-verified
//
#include <hip/hip_runtime.h>
#include <cstdint>

// Problem shape (from reference): BS=16, SQ=128, NQ=21, NK=7, D=256.
// softmax over a size-1 axis == 1.0 exactly, so:
//   attn_out[b,s,i,j,:] = keys[b,s,0,j,:]   (broadcast 21x)
//   attn_weights        = 1.0f              (everywhere)
// => pure data-movement kernel: ~308 MB NT stores + 14.7 MB reads,
//    staged through LDS via the CDNA5 async data mover for 21x reuse.

typedef float v4f __attribute__((ext_vector_type(4)));

#define BSZ 16
#define SQ  128
#define NQ  21
#define NK  7
#define DH  256

constexpr unsigned NBS     = BSZ * SQ;        // 2048 (b,s) tiles
constexpr unsigned TILE_F  = NK * DH;         // 1792 floats per key tile (7168 B)
constexpr unsigned TILE_V4 = TILE_F / 4;      // 448 16-byte chunks = blockDim (14 wave32s)
constexpr unsigned WPB     = NQ * NK;         // 147 attn_weights per (b,s) tile
constexpr unsigned long long ATTN_OUT_ELEMS =
    (unsigned long long)NBS * NQ * TILE_F;    // 77,070,336 floats

__global__ __launch_bounds__(TILE_V4)
void bahdanau_bcast_kernel(const float* __restrict__ keys, float* __restrict__ out)
{
    __shared__ __align__(16) float smem[TILE_F];

    const unsigned t  = threadIdx.x;   // 0..447 : one 16B chunk of the key tile
    const unsigned bs = blockIdx.x;    // 0..2047 : flattened (b,s)

    // ---- CDNA5 async copy: key tile (7x256 f32 = 7168 B) -> LDS ----------
    // GVS mode: mem_addr = SADDR(64b key base) + VADDR(32b offset).
    // Each of the 448 lanes moves exactly one b128 chunk; ASYNCcnt tracks it.
    unsigned lds_addr = (unsigned)(uintptr_t)smem + t * 16u;       // low 32b = LDS offset
    unsigned goff     = bs * (TILE_F * 4u) + t * 16u;              // < 2^24.. fits 32b
    asm volatile("global_load_async_to_lds_b128 %0, %1, %2"
                 :: "v"(lds_addr), "v"(goff), "s"(keys)
                 : "memory");
    asm volatile("s_wait_asynccnt 0" ::: "memory");  // this wave's async writes landed
    __syncthreads();                                 // make LDS visible across 14 waves

    // ---- broadcast 21x with streaming (non-temporal) b128 stores ---------
    // attn_out[bs][i][chunk t] = key_tile[chunk t], i = 0..20 (coalesced).
    const v4f kv = ((const v4f*)smem)[t];            // one ds_load_b128, reused 21x
    v4f* ob = (v4f*)out + (unsigned long long)bs * (NQ * TILE_V4);
#pragma unroll
    for (int i = 0; i < NQ; ++i)
        __builtin_nontemporal_store(kv, ob + i * TILE_V4 + t);

    // ---- attn_weights == 1.0f exactly (softmax over a size-1 axis) -------
    if (t < WPB)
        __builtin_nontemporal_store(
            1.0f, out + ATTN_OUT_ELEMS + (unsigned long long)bs * WPB + t);
}

extern "C" void kernel_launch(void* const* d_in, const int* /*in_sizes*/, int /*n_in*/,
                              void* d_out, int /*out_size*/, void* /*d_ws*/, size_t /*ws_size*/,
                              hipStream_t stream)
{
    // setup_inputs order: 0=queries, 1=keys, 2=cum_subs_masks, 3=num_neg,
    //                     4..9 = dead projection weights/biases (unused).
    const float* keys = (const float*)d_in[1];
    float*       out  = (float*)d_out;   // [attn_out | attn_weights], f32

    hipLaunchKernelGGL(bahdanau_bcast_kernel, dim3(NBS), dim3(TILE_V4), 0, stream,
                       keys, out);
}